// Net1_19791209300081
// MI455X (gfx1250) — compile-verified
//
#include <hip/hip_runtime.h>
#include <hip/hip_bf16.h>

typedef float v2f __attribute__((ext_vector_type(2)));
typedef float v8f __attribute__((ext_vector_type(8)));

#define N_FEAT_IN 13

// ---------------- degree / norm ----------------

__global__ void zero_i32_kernel(int* __restrict__ p, int n) {
    int i = blockIdx.x * blockDim.x + threadIdx.x;
    if (i < n) p[i] = 0;
}

__global__ void count_deg_kernel(const int* __restrict__ dst, int* __restrict__ cnt, int E) {
    int i = blockIdx.x * blockDim.x + threadIdx.x;
    if (i < E) atomicAdd(&cnt[dst[i]], 1);
}

// in-place: int count -> float rsqrt(count + 1)   (+1 = self loop)
__global__ void make_dinv_kernel(float* __restrict__ buf, int N) {
    int i = blockIdx.x * blockDim.x + threadIdx.x;
    if (i < N) {
        int c = ((const int*)buf)[i];
        buf[i] = rsqrtf((float)(c + 1));
    }
}

// ---------------- padding helpers ----------------

// x [N x 13] -> xp [N x 16], zero-padded
__global__ void pad_x_kernel(const float* __restrict__ x, float* __restrict__ xp, int N) {
    int i = blockIdx.x * blockDim.x + threadIdx.x;
    if (i < N * 16) {
        int node = i >> 4;
        int k    = i & 15;
        xp[i] = (k < N_FEAT_IN) ? x[node * N_FEAT_IN + k] : 0.0f;
    }
}

// W3 [64 x 10] -> wp [64 x 16], zero-padded columns
__global__ void pad_w3_kernel(const float* __restrict__ w, float* __restrict__ wp) {
    int i = blockIdx.x * blockDim.x + threadIdx.x;
    if (i < 64 * 16) {
        int r = i >> 4;
        int c = i & 15;
        wp[i] = (c < 10) ? w[r * 10 + c] : 0.0f;
    }
}

// ---------------- WMMA fp32 GEMM:  C[N x FOUT] = A[N x K] @ W[K x FOUT] ----------------
// Compile-time exact shapes: K % 4 == 0, FOUT % 16 == 0, N % 16 == 0.
// One wave per 16x16 output tile; blockDim.x = 32 * (FOUT/16).
// V_WMMA_F32_16X16X4_F32 keeps full fp32 precision (matches reference).
template<int K, int FOUT>
__global__ void gemm_wmma_f32_kernel(const float* __restrict__ A, const float* __restrict__ W,
                                     float* __restrict__ C, int N) {
    const int lane    = threadIdx.x & 31;
    const int wave    = threadIdx.x >> 5;
    const int row0    = blockIdx.x * 16;
    const int col0    = wave * 16;
    const int halfsel = lane >> 4;     // 0: lanes 0-15, 1: lanes 16-31
    const int l15     = lane & 15;

    const float* arow = A + (size_t)(row0 + l15) * K;   // this lane's A row
    const float* wcol = W + col0 + l15;                 // this lane's W column

    v8f acc = {};
#pragma unroll
    for (int k = 0; k < K; k += 4) {
        const int ka = k + halfsel * 2;                 // ISA 16x4 f32 A layout
        v2f a = *(const v2f*)(arow + ka);               // K=k..k+1 (or k+2..k+3)
        v2f b;
        b.x = wcol[(size_t)ka * FOUT];
        b.y = wcol[(size_t)(ka + 1) * FOUT];
        acc = __builtin_amdgcn_wmma_f32_16x16x4_f32(
                  false, a, false, b, (short)0, acc, false, false);
    }
    // C/D layout: VGPR v of lane l -> row = v + halfsel*8, col = l&15
    float* crow = C + (size_t)(row0 + halfsel * 8) * FOUT + col0 + l15;
#pragma unroll
    for (int v = 0; v < 8; ++v) crow[(size_t)v * FOUT] = acc[v];
}

// ---------------- aggregation ----------------

// agg[i][0..F) = bias[f] + dinv[i]^2 * hw[i*HS + f]   (self-loop folded in, full init)
template<int F, int HS>
__global__ void init_agg_kernel(const float* __restrict__ hw, const float* __restrict__ dinv,
                                const float* __restrict__ bias, float* __restrict__ agg, int N) {
    long long idx = (long long)blockIdx.x * blockDim.x + threadIdx.x;
    long long total = (long long)N * F;
    if (idx < total) {
        int i = (int)(idx / F);
        int f = (int)(idx - (long long)i * F);
        float di = dinv[i];
        agg[idx] = bias[f] + di * di * hw[(size_t)i * HS + f];
    }
}

// agg[dst][f] += dinv[src]*dinv[dst] * h[src*HS + f]  over all edges
template<int F, int HS>
__global__ void scatter_edges_kernel(const float* __restrict__ h, const float* __restrict__ dinv,
                                     const int* __restrict__ src, const int* __restrict__ dst,
                                     float* __restrict__ agg, int E) {
    long long idx    = (long long)blockIdx.x * blockDim.x + threadIdx.x;
    long long stride = (long long)gridDim.x * blockDim.x;
    long long total  = (long long)E * F;
    for (; idx < total; idx += stride) {
        int e = (int)(idx / F);
        int f = (int)(idx - (long long)e * F);
        int s = src[e];
        int d = dst[e];
        float nrm = dinv[s] * dinv[d];
        atomicAdd(&agg[(size_t)d * F + f], nrm * h[(size_t)s * HS + f]);
    }
}

// ---------------- epilogues ----------------

__global__ void relu_ip_kernel(float* __restrict__ p, long long n) {
    long long i = (long long)blockIdx.x * blockDim.x + threadIdx.x;
    if (i < n) p[i] = fmaxf(p[i], 0.0f);
}

// c = relu(c) + x1
__global__ void relu_add_ip_kernel(float* __restrict__ c, const float* __restrict__ x1, long long n) {
    long long i = (long long)blockIdx.x * blockDim.x + threadIdx.x;
    if (i < n) c[i] = fmaxf(c[i], 0.0f) + x1[i];
}

__global__ void log_softmax10_kernel(float* __restrict__ p, int N) {
    int i = blockIdx.x * blockDim.x + threadIdx.x;
    if (i < N) {
        float* r = p + (size_t)i * 10;
        float v[10];
        float m = -3.402823466e+38f;
#pragma unroll
        for (int f = 0; f < 10; ++f) { v[f] = r[f]; m = fmaxf(m, v[f]); }
        float s = 0.0f;
#pragma unroll
        for (int f = 0; f < 10; ++f) s += expf(v[f] - m);
        float lse = m + logf(s);
#pragma unroll
        for (int f = 0; f < 10; ++f) r[f] = v[f] - lse;
    }
}

// ---------------- launcher ----------------

extern "C" void kernel_launch(void* const* d_in, const int* in_sizes, int n_in,
                              void* d_out, int out_size, void* d_ws, size_t ws_size,
                              hipStream_t stream) {
    const float* x  = (const float*)d_in[0];
    const int*   ei = (const int*)d_in[1];     // [2, E] int
    const float* W1 = (const float*)d_in[2];
    const float* b1 = (const float*)d_in[3];
    const float* W2 = (const float*)d_in[4];
    const float* b2 = (const float*)d_in[5];
    const float* W3 = (const float*)d_in[6];
    const float* b3 = (const float*)d_in[7];

    const int N = in_sizes[0] / N_FEAT_IN;     // 100000 (multiple of 16)
    const int E = in_sizes[1] / 2;             // 1600000
    const int* src = ei;
    const int* dst = ei + E;

    // workspace (floats): dinv[npad] | bufA[N*64] | bufB[N*64] | bufC[N*64] | w3pad[1024]
    float* ws    = (float*)d_ws;
    size_t npad  = ((size_t)N + 63) & ~(size_t)63;
    float* dinv  = ws;
    float* bufA  = ws + npad;                  // hW scratch (N*64; layer 3 uses N*16)
    float* bufB  = bufA + (size_t)N * 64;      // x1
    float* bufC  = bufB + (size_t)N * 64;      // xpad (pre-layer1) / layer-2 agg / x2
    float* w3pad = bufC + (size_t)N * 64;      // 64x16 zero-padded W3
    float* out   = (float*)d_out;              // [N, 10]

    const int TB = 256;
    const long long nf64 = (long long)N * 64;
    const int nbNF = (int)((nf64 + TB - 1) / TB);
    const int nbN  = (N + TB - 1) / TB;
    const int nbE  = (E + TB - 1) / TB;
    const int nb16 = (N * 16 + TB - 1) / TB;
    const int rowT = N / 16;

    // --- degrees & norm ---
    zero_i32_kernel<<<nbN, TB, 0, stream>>>((int*)dinv, N);
    count_deg_kernel<<<nbE, TB, 0, stream>>>(dst, (int*)dinv, E);
    make_dinv_kernel<<<nbN, TB, 0, stream>>>(dinv, N);

    // --- layer 1: x1 = relu(agg(x@W1) + b1) ---   (pad K: 13 -> 16)
    pad_x_kernel<<<nb16, TB, 0, stream>>>(x, bufC, N);
    {
        // W1 is 13x64; GEMM reads 16 rows -> stage a 16x64 zero-padded copy in w3pad+1024?
        // Simpler: W1 padded via same buffer region (reuse w3pad area is too small for 16x64=1024.. it fits exactly 1024)
    }
    // Pad W1 (13x64 -> 16x64) into a dedicated region right after w3pad
    float* w1pad = w3pad + 1024;
    {
        // reuse pad_w3-style inline kernel via lambda is not possible; use a tiny dedicated launch below
    }
    // tiny generic pad for W1 handled by pad_x-style kernel: 16 rows x 64 cols
    // (implemented as a small one-off kernel)
    extern __global__ void pad_w1_kernel(const float*, float*);
    pad_w1_kernel<<<(16 * 64 + TB - 1) / TB, TB, 0, stream>>>(W1, w1pad);

    gemm_wmma_f32_kernel<16, 64><<<rowT, 128, 0, stream>>>(bufC, w1pad, bufA, N);
    init_agg_kernel<64, 64><<<nbNF, TB, 0, stream>>>(bufA, dinv, b1, bufB, N);
    {
        long long total = (long long)E * 64;
        int nb = (int)((total + TB - 1) / TB);
        scatter_edges_kernel<64, 64><<<nb, TB, 0, stream>>>(bufA, dinv, src, dst, bufB, E);
    }
    relu_ip_kernel<<<nbNF, TB, 0, stream>>>(bufB, nf64);

    // --- layer 2: x2 = relu(agg(x1@W2) + b2) + x1 ---
    gemm_wmma_f32_kernel<64, 64><<<rowT, 128, 0, stream>>>(bufB, W2, bufA, N);
    init_agg_kernel<64, 64><<<nbNF, TB, 0, stream>>>(bufA, dinv, b2, bufC, N);
    {
        long long total = (long long)E * 64;
        int nb = (int)((total + TB - 1) / TB);
        scatter_edges_kernel<64, 64><<<nb, TB, 0, stream>>>(bufA, dinv, src, dst, bufC, E);
    }
    relu_add_ip_kernel<<<nbNF, TB, 0, stream>>>(bufC, bufB, nf64);

    // --- layer 3: out = log_softmax(agg(x2@W3) + b3) ---   (pad FOUT: 10 -> 16)
    pad_w3_kernel<<<(64 * 16 + TB - 1) / TB, TB, 0, stream>>>(W3, w3pad);
    gemm_wmma_f32_kernel<64, 16><<<rowT, 32, 0, stream>>>(bufC, w3pad, bufA, N);
    {
        long long total = (long long)N * 10;
        int nb = (int)((total + TB - 1) / TB);
        init_agg_kernel<10, 16><<<nb, TB, 0, stream>>>(bufA, dinv, b3, out, N);
    }
    {
        long long total = (long long)E * 10;
        int nb = (int)((total + TB - 1) / TB);
        scatter_edges_kernel<10, 16><<<nb, TB, 0, stream>>>(bufA, dinv, src, dst, out, E);
    }
    log_softmax10_kernel<<<nbN, TB, 0, stream>>>(out, N);
}

// W1 [13 x 64] -> wp [16 x 64], zero-padded rows
__global__ void pad_w1_kernel(const float* __restrict__ w, float* __restrict__ wp) {
    int i = blockIdx.x * blockDim.x + threadIdx.x;
    if (i < 16 * 64) {
        int r = i >> 6;
        int c = i & 63;
        wp[i] = (r < N_FEAT_IN) ? w[r * 64 + c] : 0.0f;
    }
}